// GSeparableKernel_66048007078602
// MI455X (gfx1250) — compile-verified
//
#include <hip/hip_runtime.h>
#include <hip/hip_bf16.h>

typedef __attribute__((ext_vector_type(2))) float v2f;
typedef __attribute__((ext_vector_type(8))) float v8f;

#define O_CH   256
#define C_IN   128
#define K_H    24
#define N_IN   24
#define N_OUT  24
#define KS     5
#define WH_ELEMS (O_CH * N_IN * C_IN * N_OUT)   // 18,874,368
#define SEL_STRIDE 32                            // padded N for S
#define SEL_PER_A  (K_H * SEL_STRIDE)           // 768 floats per 'a'

// ---------------------------------------------------------------------------
// Kernel 1: nn(a,b) argmax on SO(3)  +  build one-hot selection tensor
// S[a][h][b_pad32] = (nn(a,b) == h) ? 1 : 0     (b cols 24..31 stay zero)
// ---------------------------------------------------------------------------
__global__ void nn_select_kernel(const float* __restrict__ in_H,
                                 const float* __restrict__ out_H,
                                 const float* __restrict__ grid_H,
                                 float* __restrict__ S) {
    const int t = threadIdx.x;                       // 0..575
    for (int i = t; i < N_IN * SEL_PER_A; i += N_IN * N_OUT)
        S[i] = 0.0f;
    __syncthreads();

    const int a = t / N_OUT;
    const int b = t % N_OUT;
    const float* A = in_H  + a * 9;
    const float* B = out_H + b * 9;

    // H[i][k] = sum_j out_H[b][j][i] * in_H[a][j][k]   (out_inv = out_H^T)
    float H[9];
#pragma unroll
    for (int i = 0; i < 3; ++i)
#pragma unroll
        for (int k = 0; k < 3; ++k)
            H[i * 3 + k] = B[0 * 3 + i] * A[0 * 3 + k]
                         + B[1 * 3 + i] * A[1 * 3 + k]
                         + B[2 * 3 + i] * A[2 * 3 + k];

    float best = -1e30f;
    int   bi   = 0;
    for (int h = 0; h < K_H; ++h) {
        const float* G = grid_H + h * 9;
        float s = 0.0f;
#pragma unroll
        for (int e = 0; e < 9; ++e) s += H[e] * G[e];
        if (s > best) { best = s; bi = h; }          // first-max, like jnp.argmax
    }
    S[a * SEL_PER_A + bi * SEL_STRIDE + b] = 1.0f;
}

// ---------------------------------------------------------------------------
// Kernel 2: wH[o,a,c,b] = weight_H[o,c,nn(a,b)] as 24 GEMMs with one-hot S,
// via V_WMMA_F32_16X16X4_F32. One wave per 16-row M-tile.
//   A-frag (16x4):  lane l -> M = l%16,          K = 4k + 2*(l/16) + vgpr
//   B-frag (4x16):  lane l -> N = l%16,          K = 4k + 2*(l/16) + vgpr
//   D      (16x16): lane l -> M = vgpr+8*(l/16), N = l%16
// Epilogue: stage 16x24 D tile in LDS, then 3x global_store_b128 per lane
// (each wave's 16 rows are 16 consecutive 'c' rows -> 1536 contiguous bytes).
// ---------------------------------------------------------------------------
__global__ __launch_bounds__(256) void
wh_wmma_kernel(const float* __restrict__ W,     // weight_H [32768 x 24]
               const float* __restrict__ S,     // [24][24][32]
               float* __restrict__ out) {       // wH [256,24,128,24]
    __shared__ __align__(16) float tile[8][16 * N_OUT];   // 12 KB / block

    const int a     = blockIdx.x >> 8;          // / 256
    const int blk   = blockIdx.x & 255;
    const int wave  = threadIdx.x >> 5;
    const int lane  = threadIdx.x & 31;
    const int mtile = blk * 8 + wave;           // 0..2047
    const int half  = lane >> 4;                // 0 or 1
    const int l     = lane & 15;

    const float* Arow = W + (mtile * 16 + l) * K_H;   // one 24-float row
    const float* Sa   = S + a * SEL_PER_A;

    v8f c0 = {0.f,0.f,0.f,0.f,0.f,0.f,0.f,0.f};
    v8f c1 = c0;

#pragma unroll
    for (int k = 0; k < 6; ++k) {               // K = 24 = 6 steps of 4
        const int kk = 4 * k + 2 * half;
        v2f av = *(const v2f*)(Arow + kk);      // 8B-aligned pair
        v2f b0, b1;
        b0.x = Sa[kk * SEL_STRIDE + l];         b0.y = Sa[(kk + 1) * SEL_STRIDE + l];
        b1.x = Sa[kk * SEL_STRIDE + 16 + l];    b1.y = Sa[(kk + 1) * SEL_STRIDE + 16 + l];
        c0 = __builtin_amdgcn_wmma_f32_16x16x4_f32(false, av, false, b0,
                                                   (short)0, c0, false, false);
        c1 = __builtin_amdgcn_wmma_f32_16x16x4_f32(false, av, false, b1,
                                                   (short)0, c1, false, false);
    }

    // D fragments -> per-wave LDS tile [16 rows][24 cols]
    float* t = tile[wave];
#pragma unroll
    for (int v = 0; v < 8; ++v) {
        const int M = v + 8 * half;
        t[M * N_OUT + l] = c0[v];               // N-tile 0: b = 0..15
        if (l < 8)
            t[M * N_OUT + 16 + l] = c1[v];      // N-tile 1: b = 16..23
    }
    __syncthreads();

    // Contiguous 1536-byte tile -> wide coalesced stores
    const int o    = mtile >> 3;                // mg/128 = mtile*16/128
    const int cch0 = (mtile & 7) * 16;
    float* base = out + ((size_t)((o * N_IN + a) * C_IN + cch0)) * N_OUT;
    const float4* src = (const float4*)t;       // 96 x float4
    float4*       dst = (float4*)base;
#pragma unroll
    for (int i = 0; i < 3; ++i)
        dst[lane + 32 * i] = src[lane + 32 * i];
}

// ---------------------------------------------------------------------------
// Kernel 3: wRn[o,b,:] = trilinear(weight_Rn[o], out_H[b]^T @ grid_Rn)
// Exact align_corners=True + zero-padding semantics of the reference.
// ---------------------------------------------------------------------------
__global__ void wrn_kernel(const float* __restrict__ out_H,
                           const float* __restrict__ grid_Rn,
                           const float* __restrict__ weight_Rn,
                           float* __restrict__ out) {
    const int o = blockIdx.x / N_OUT;
    const int b = blockIdx.x % N_OUT;
    const int t = threadIdx.x;

    __shared__ float w[KS * KS * KS];
    if (t < KS * KS * KS) w[t] = weight_Rn[o * (KS * KS * KS) + t];
    __syncthreads();
    if (t >= KS * KS * KS) return;

    const float g0 = grid_Rn[0 * 125 + t];
    const float g1 = grid_Rn[1 * 125 + t];
    const float g2 = grid_Rn[2 * 125 + t];
    const float* R = out_H + b * 9;
    // coord_i = sum_j out_H[b][j][i] * g_j
    const float c0 = R[0] * g0 + R[3] * g1 + R[6] * g2;
    const float c1 = R[1] * g0 + R[4] * g1 + R[7] * g2;
    const float c2 = R[2] * g0 + R[5] * g1 + R[8] * g2;

    const float cz = (c0 + 1.0f) * 0.5f * (KS - 1);
    const float cy = (c1 + 1.0f) * 0.5f * (KS - 1);
    const float cx = (c2 + 1.0f) * 0.5f * (KS - 1);
    const float z0f = floorf(cz), y0f = floorf(cy), x0f = floorf(cx);
    const int z0 = (int)z0f, y0 = (int)y0f, x0 = (int)x0f;
    const float fz = cz - z0f, fy = cy - y0f, fx = cx - x0f;

    auto g = [&](int iz, int iy, int ix) -> float {
        const bool valid = (iz >= 0) & (iz < KS) & (iy >= 0) & (iy < KS)
                         & (ix >= 0) & (ix < KS);
        const int zz = min(max(iz, 0), KS - 1);
        const int yy = min(max(iy, 0), KS - 1);
        const int xx = min(max(ix, 0), KS - 1);
        const float v = w[zz * 25 + yy * 5 + xx];
        return valid ? v : 0.0f;
    };

    const float val =
          g(z0,     y0,     x0    ) * (1 - fz) * (1 - fy) * (1 - fx)
        + g(z0,     y0,     x0 + 1) * (1 - fz) * (1 - fy) * fx
        + g(z0,     y0 + 1, x0    ) * (1 - fz) * fy       * (1 - fx)
        + g(z0,     y0 + 1, x0 + 1) * (1 - fz) * fy       * fx
        + g(z0 + 1, y0,     x0    ) * fz       * (1 - fy) * (1 - fx)
        + g(z0 + 1, y0,     x0 + 1) * fz       * (1 - fy) * fx
        + g(z0 + 1, y0 + 1, x0    ) * fz       * fy       * (1 - fx)
        + g(z0 + 1, y0 + 1, x0 + 1) * fz       * fy       * fx;

    out[((size_t)(o * N_OUT + b)) * 125 + t] = val;
}

extern "C" void kernel_launch(void* const* d_in, const int* in_sizes, int n_in,
                              void* d_out, int out_size, void* d_ws, size_t ws_size,
                              hipStream_t stream) {
    const float* in_H      = (const float*)d_in[0];
    const float* out_H     = (const float*)d_in[1];
    const float* grid_H    = (const float*)d_in[2];
    const float* grid_Rn   = (const float*)d_in[3];
    const float* weight_H  = (const float*)d_in[4];
    const float* weight_Rn = (const float*)d_in[5];

    float* wH  = (float*)d_out;                  // [256,24,128,24]
    float* wRn = (float*)d_out + WH_ELEMS;       // [256,24,1,5,5,5]
    float* S   = (float*)d_ws;                   // 24*24*32 floats = 72 KB

    // 1) nn argmax + one-hot selection matrices
    nn_select_kernel<<<1, N_IN * N_OUT, 0, stream>>>(in_H, out_H, grid_H, S);
    // 2) wH broadcast as 24 one-hot GEMMs via f32 WMMA
    wh_wmma_kernel<<<N_IN * 256, 256, 0, stream>>>(weight_H, S, wH);
    // 3) wRn trilinear resample
    wrn_kernel<<<O_CH * N_OUT, 128, 0, stream>>>(out_H, grid_Rn, weight_Rn, wRn);
}